// NPSLoss_2422361555121
// MI455X (gfx1250) — compile-verified
//
#include <hip/hip_runtime.h>
#include <math.h>

// NPS loss: mean over pixels of min_c ||p - c||, p from (3,1024,2048) fp32
// reshaped (-1,3), 30 colors. dist^2 = ||p||^2 - 2 p.c + ||c||^2.
// Colors -> A (16x4 f32, loop-invariant), pixels -> B (4x16),
// V_WMMA_F32_16X16X4_F32 gives -2p.c + ||c||^2 for 16 colors x 16 pixels.
// Pixels on the N axis => min-over-colors is a per-lane VGPR tree + one
// v_permlanex16 half-swap (no LDS traffic).

typedef __attribute__((ext_vector_type(2))) float v2f;
typedef __attribute__((ext_vector_type(8))) float v8f;

// xor-16 half swap via v_permlanex16_b32 (identity selects) — pure VALU.
__device__ __forceinline__ float xor16f(float x) {
    int i = __float_as_int(x);
    int r = __builtin_amdgcn_permlanex16(i, i, 0x76543210, 0xfedcba98, false, false);
    return __int_as_float(r);
}

__global__ void __launch_bounds__(256)
nps_main(const float* __restrict__ pix,   // nPix*3 floats
         const float* __restrict__ cols,  // nCol*3 floats
         float* __restrict__ partial,     // one float per block
         int nPix, int nCol, int numTiles)
{
    const int lane = (int)(threadIdx.x & 31);
    const int half = (lane >> 4) & 1;   // 0 = lanes 0-15, 1 = lanes 16-31
    const int m0   = lane & 15;

    // ---- A matrices (colors), constant across the tile loop ----
    // A row m = [-2cx, -2cy, -2cz, ||c||^2]; 16x4 f32 A layout:
    // lanes 0-15 hold K=0,1 in the two VGPRs; lanes 16-31 hold K=2,3.
    v2f a0, a1;
    {
        const float BIG = 1.0e30f;
        int c0 = m0;        // tile 0: colors 0..15
        int c1 = 16 + m0;   // tile 1: colors 16..31 (padded)
        float x0=0.f,y0=0.f,z0=0.f, x1=0.f,y1=0.f,z1=0.f;
        if (c0 < nCol) { x0 = cols[3*c0]; y0 = cols[3*c0+1]; z0 = cols[3*c0+2]; }
        if (c1 < nCol) { x1 = cols[3*c1]; y1 = cols[3*c1+1]; z1 = cols[3*c1+2]; }
        float n0 = x0*x0 + y0*y0 + z0*z0;
        float n1 = (c1 < nCol) ? (x1*x1 + y1*y1 + z1*z1) : BIG; // pad -> huge dist
        if (half == 0) {
            a0.x = -2.0f*x0; a0.y = -2.0f*y0;   // K=0,1 rows
            a1.x = -2.0f*x1; a1.y = -2.0f*y1;
        } else {
            a0.x = -2.0f*z0; a0.y = n0;          // K=2,3 rows
            a1.x = -2.0f*z1; a1.y = n1;
        }
    }

    const v8f czero = {};
    float acc = 0.0f;

    // Wave-uniform (scalar) tile loop: force the induction into SGPRs so the
    // compiler emits s_cbranch with EXEC all-1s (required around WMMA).
    const int wlocal = __builtin_amdgcn_readfirstlane((int)(threadIdx.x >> 5));
    const int nwpb   = (int)(blockDim.x >> 5);
    const int gwave  = (int)blockIdx.x * nwpb + wlocal;
    const int nwaves = (int)gridDim.x * nwpb;

    for (int tile = gwave; tile < numTiles; tile += nwaves) {
        int p      = tile * 16 + m0;
        bool valid = (p < nPix);
        int  pc    = valid ? p : 0;              // keep loads in-bounds

        // half0 loads floats [3p, 3p+1] = {x,y}; half1 loads [3p+1, 3p+2] = {y,z}
        const float* base = pix + 3*pc + half;
        float f0 = __builtin_nontemporal_load(base);
        float f1 = __builtin_nontemporal_load(base + 1);

        // B column n = [x, y, z, 1]; lane n: {x,y}, lane n+16: {z, 1}  (branchless)
        v2f b;
        b.x = half ? f1 : f0;
        b.y = half ? 1.0f : f1;
        float e = half ? 0.0f : f1;              // extra term for ||p||^2 partial
        float s = b.x*b.x + e*f1;                // half0: x^2+y^2, half1: z^2

        v8f d0 = __builtin_amdgcn_wmma_f32_16x16x4_f32(
                     false, a0, false, b, (short)0, czero, false, false);
        v8f d1 = __builtin_amdgcn_wmma_f32_16x16x4_f32(
                     false, a1, false, b, (short)0, czero, false, false);

        // min over the 16 color results held in this lane's D VGPRs
        float m = fminf(d0[0], d1[0]);
#pragma unroll
        for (int i = 1; i < 8; ++i) m = fminf(m, fminf(d0[i], d1[i]));
        // merge with partner half (the other 16 colors for this pixel)
        m = fminf(m, xor16f(m));

        // ||p||^2 = (x^2+y^2) + z^2 combined across the lane pair
        float np2 = s + xor16f(s);

        // raw v_sqrt_f32; clamp tiny negative rounding before sqrt
        float t = __builtin_amdgcn_sqrtf(fmaxf(m + np2, 0.0f));
        acc += valid ? t : 0.0f;                 // pixel counted by both halves
    }

    // deterministic in-wave butterfly sum (cold path; shfl is fine here)
#pragma unroll
    for (int off = 16; off >= 1; off >>= 1) acc += __shfl_xor(acc, off);

    __shared__ float wsum[8];                    // 256 threads = 8 waves
    int wid = (int)(threadIdx.x >> 5);
    if (lane == 0) wsum[wid] = acc;
    __syncthreads();
    if (threadIdx.x == 0) {
        float bsum = 0.0f;
        int nw = (int)(blockDim.x >> 5);
        for (int i = 0; i < nw; ++i) bsum += wsum[i]; // fixed order
        partial[blockIdx.x] = bsum;
    }
}

__global__ void __launch_bounds__(256)
nps_reduce(const float* __restrict__ partial, int nPartial,
           float* __restrict__ out, float scale)
{
    __shared__ float sm[256];
    float s = 0.0f;
    for (int i = (int)threadIdx.x; i < nPartial; i += 256) s += partial[i]; // fixed order
    sm[threadIdx.x] = s;
    __syncthreads();
#pragma unroll
    for (int st = 128; st > 0; st >>= 1) {       // fixed pairing -> deterministic
        if ((int)threadIdx.x < st) sm[threadIdx.x] += sm[threadIdx.x + st];
        __syncthreads();
    }
    if (threadIdx.x == 0) out[0] = sm[0] * scale;
}

extern "C" void kernel_launch(void* const* d_in, const int* in_sizes, int n_in,
                              void* d_out, int out_size, void* d_ws, size_t ws_size,
                              hipStream_t stream)
{
    const float* pix  = (const float*)d_in[0];   // adv_patch, 3*1024*2048 fp32
    const float* cols = (const float*)d_in[1];   // printable_colors, 30*3 fp32

    const int nPix     = in_sizes[0] / 3;        // 2,097,152
    const int nCol     = in_sizes[1] / 3;        // 30
    const int numTiles = (nPix + 15) / 16;

    const int BLOCKS = 1024;                     // 4 KB of d_ws for partials
    const int TPB    = 256;                      // 8 waves/block (wave32)

    float* partial = (float*)d_ws;

    nps_main<<<BLOCKS, TPB, 0, stream>>>(pix, cols, partial, nPix, nCol, numTiles);

    const float COEFFICIENT = 1.0f;
    // each pixel's sqrt is accumulated by both lane halves -> 0.5 factor
    const float scale = 0.5f * COEFFICIENT / (float)nPix;
    nps_reduce<<<1, 256, 0, stream>>>(partial, BLOCKS, (float*)d_out, scale);
}